// GroupConnected_39685497815124
// MI455X (gfx1250) — compile-verified
//
#include <hip/hip_runtime.h>

// Problem shape (fixed by the reference).
#define B_DIM 8192
#define F_DIM 8192
#define G_DIM 2048
#define K_DIM 8

#define ROWS    4     // rows of x staged in LDS per block (4 * 32KB = 128KB LDS)
#define THREADS 256   // 8 wave32 waves per block
#define CHUNKS  ((ROWS * F_DIM) / (THREADS * 4))   // 32 b128 async copies per thread

#define GLOBAL_AS __attribute__((address_space(1)))
#define LDS_AS    __attribute__((address_space(3)))

typedef int v4i __attribute__((ext_vector_type(4)));

// out[b,g] = sum_k x[b, idx[g,k]] * w[g,k]
// Strategy: stage ROWS rows of x in LDS via CDNA5 async global->LDS DMA,
// then do all G*K gathers out of LDS. HBM traffic = read x once + write out
// once (~320 MB total -> ~14 us roofline at 23.3 TB/s).
__global__ __launch_bounds__(THREADS)
void GroupConnected_gather_kernel(const float* __restrict__ x,
                                  const int*   __restrict__ gidx,
                                  const float* __restrict__ w,
                                  float*       __restrict__ out)
{
    __shared__ float rows[ROWS * F_DIM];   // 128 KB of the WGP's 320 KB LDS

    const int  tid = threadIdx.x;
    const long b0  = (long)blockIdx.x * ROWS;
    const float* src = x + b0 * (long)F_DIM;

    // ---- Stage ROWS contiguous rows of x into LDS (async, ASYNCcnt-tracked).
    // Fully unrolled: 32 back-to-back b128 async DMA ops per thread, each
    // advancing by THREADS*4 floats (4 KB). No loop masking, no branches.
    {
        const float* gp = src  + tid * 4;
        float*       lp = rows + tid * 4;
#pragma unroll
        for (int j = 0; j < CHUNKS; ++j) {
            __builtin_amdgcn_global_load_async_to_lds_b128(
                (GLOBAL_AS v4i*)(gp + j * (THREADS * 4)),
                (LDS_AS    v4i*)(lp + j * (THREADS * 4)),
                /*offset=*/0, /*cpol=*/0);
        }
    }
    __builtin_amdgcn_s_wait_asynccnt(0);   // this wave's async copies landed in LDS
    __syncthreads();                        // all waves' copies landed

    // ---- Gather + dot phase: each thread handles G_DIM/THREADS = 8 groups,
    // producing ROWS outputs per group.
#pragma unroll 2
    for (int g = tid; g < G_DIM; g += THREADS) {
        const int4*   ip = (const int4*)(gidx + g * K_DIM);
        const float4* wp = (const float4*)(w    + g * K_DIM);
        const int4   i0 = ip[0], i1 = ip[1];
        const float4 w0 = wp[0], w1 = wp[1];

        const int   idx[K_DIM] = {i0.x, i0.y, i0.z, i0.w, i1.x, i1.y, i1.z, i1.w};
        const float wv [K_DIM] = {w0.x, w0.y, w0.z, w0.w, w1.x, w1.y, w1.z, w1.w};

        float acc0 = 0.f, acc1 = 0.f, acc2 = 0.f, acc3 = 0.f;
#pragma unroll
        for (int k = 0; k < K_DIM; ++k) {
            const float wk = wv[k];
            const int   ik = idx[k];
            acc0 = __builtin_fmaf(rows[0 * F_DIM + ik], wk, acc0);
            acc1 = __builtin_fmaf(rows[1 * F_DIM + ik], wk, acc1);
            acc2 = __builtin_fmaf(rows[2 * F_DIM + ik], wk, acc2);
            acc3 = __builtin_fmaf(rows[3 * F_DIM + ik], wk, acc3);
        }

        // Output is write-once: bypass caches with non-temporal stores.
        __builtin_nontemporal_store(acc0, &out[(b0 + 0) * G_DIM + g]);
        __builtin_nontemporal_store(acc1, &out[(b0 + 1) * G_DIM + g]);
        __builtin_nontemporal_store(acc2, &out[(b0 + 2) * G_DIM + g]);
        __builtin_nontemporal_store(acc3, &out[(b0 + 3) * G_DIM + g]);
    }
}

extern "C" void kernel_launch(void* const* d_in, const int* in_sizes, int n_in,
                              void* d_out, int out_size, void* d_ws, size_t ws_size,
                              hipStream_t stream)
{
    (void)in_sizes; (void)n_in; (void)out_size; (void)d_ws; (void)ws_size;

    const float* x    = (const float*)d_in[0];   // (B, F) fp32
    const int*   gidx = (const int*)  d_in[1];   // (G, K) int
    const float* w    = (const float*)d_in[2];   // (G, K) fp32
    float*       out  = (float*)d_out;           // (B, G) fp32

    dim3 grid(B_DIM / ROWS);   // 2048 blocks
    dim3 block(THREADS);       // 256 threads = 8 wave32
    GroupConnected_gather_kernel<<<grid, block, 0, stream>>>(x, gidx, w, out);
}